// PoseEstimator_21474836480057
// MI455X (gfx1250) — compile-verified
//
#include <hip/hip_runtime.h>
#include <math.h>

// ---------------------------------------------------------------------------
// PoseEstimator chamfer objective on gfx1250 (MI455X).
// d2[m,n] = |X_m|^2 + |y_n|^2 - 2 X_m . y_n  computed with
// V_WMMA_F32_16X16X4_F32:  A[m,:] = [-2X0,-2X1,-2X2, 1]  (16x4)
//                          B[:,n] = [y0, y1, y2, |y|^2]   (4x16)
//                          C[m,n] = |X_m|^2 (preloaded accumulator)
// One WMMA = 256 full squared distances. Two passes (roles swapped) give
// row-mins in both directions with no atomics.
//
// B-side points are pre-swizzled by the prep kernels into the exact WMMA
// B-fragment lane layout (16-point group -> 32 lane slots x 8B), so the hot
// loop is just: ds_load_b64 -> v_wmma -> v_min3 accumulation.
// ---------------------------------------------------------------------------

typedef float v2f __attribute__((ext_vector_type(2)));
typedef float v8f __attribute__((ext_vector_type(8)));

#define PARTS 4
#define NPTS  8192
#define ROWS_PER_BLOCK 256   // 16 waves x 16 rows
#define BCHUNK 2048          // points of B staged in LDS per chunk (32 KB)

__device__ __forceinline__ void quat_to_rt(const float* __restrict__ q4,
                                           const float* __restrict__ t3,
                                           float R[9], float T[3]) {
    float a = q4[0], b = q4[1], c = q4[2], d = q4[3];
    float inv = rsqrtf(a*a + b*b + c*c + d*d);
    a *= inv; b *= inv; c *= inv; d *= inv;
    R[0] = 1.f - 2.f*c*c - 2.f*d*d;  R[1] = 2.f*b*c - 2.f*a*d;  R[2] = 2.f*a*c + 2.f*b*d;
    R[3] = 2.f*b*c + 2.f*a*d;        R[4] = 1.f - 2.f*b*b - 2.f*d*d;  R[5] = 2.f*c*d - 2.f*a*b;
    R[6] = 2.f*b*d - 2.f*a*c;        R[7] = 2.f*a*b + 2.f*c*d;  R[8] = 1.f - 2.f*b*b - 2.f*c*c;
    T[0] = t3[0]; T[1] = t3[1]; T[2] = t3[2];
}

// d_out[1..64] = transforms (P,4,4) row-major
__global__ void write_transforms_kernel(const float* __restrict__ quat,
                                        const float* __restrict__ tra,
                                        float* __restrict__ out) {
    int t = threadIdx.x;
    if (t >= PARTS * 16) return;
    int p = t >> 4, i = (t >> 2) & 3, j = t & 3;
    float R[9], T[3];
    quat_to_rt(quat + 4*p, tra + 3*p, R, T);
    float v;
    if (i < 3) v = (j < 3) ? R[i*3 + j] : T[i];
    else       v = (j == 3) ? 1.f : 0.f;
    out[t] = v;
}

// Write both representations of a prepared point:
//   pt4[idx]  = (x, y, z, n2)                       (A-role: rows + C norms)
//   bfrag     = WMMA B-fragment lane layout         (B-role)
// B-fragment layout per 16-point group g (32 float2 slots):
//   slot l      (lanes 0-15)  = (x_l, y_l)      -> K = {0,1}
//   slot 16 + l (lanes 16-31) = (z_l, n2_l)     -> K = {2,3}
__device__ __forceinline__ void store_prepared(float4* __restrict__ pt4,
                                               float2* __restrict__ bfrag,
                                               size_t base, int idx,
                                               float x, float y, float z, float n2) {
    pt4[base + idx] = make_float4(x, y, z, n2);
    int g = idx >> 4, l = idx & 15;
    float2* fb = bfrag + (base + (size_t)g * 16) * 2;   // 32 float2 per group
    fb[l]      = make_float2(x, y);
    fb[16 + l] = make_float2(z, n2);
}

// cadT: X = R*cad + t
__global__ void prep_cad_kernel(const float* __restrict__ cad,
                                const float* __restrict__ quat,
                                const float* __restrict__ tra,
                                float4* __restrict__ pt4,
                                float2* __restrict__ bfrag) {
    int p   = blockIdx.x;
    int idx = blockIdx.y * blockDim.x + threadIdx.x;
    float R[9], T[3];
    quat_to_rt(quat + 4*p, tra + 3*p, R, T);
    const float* c = cad + ((size_t)p * NPTS + idx) * 3;
    float x = c[0], y = c[1], z = c[2];
    float X = R[0]*x + R[1]*y + R[2]*z + T[0];
    float Y = R[3]*x + R[4]*y + R[5]*z + T[1];
    float Z = R[6]*x + R[7]*y + R[8]*z + T[2];
    store_prepared(pt4, bfrag, (size_t)p * NPTS, idx, X, Y, Z, X*X + Y*Y + Z*Z);
}

__global__ void prep_cam_kernel(const float* __restrict__ cam,
                                float4* __restrict__ pt4,
                                float2* __restrict__ bfrag) {
    int p   = blockIdx.x;
    int idx = blockIdx.y * blockDim.x + threadIdx.x;
    const float* c = cam + ((size_t)p * NPTS + idx) * 3;
    float x = c[0], y = c[1], z = c[2];
    store_prepared(pt4, bfrag, (size_t)p * NPTS, idx, x, y, z, x*x + y*y + z*z);
}

// For each A-row, min over all B-points of the squared distance.
// grid = (PARTS, NPTS/ROWS_PER_BLOCK), block = 512 (16 waves).
__global__ void chamfer_rowmin_kernel(const float4* __restrict__ Apts,
                                      const float2* __restrict__ Bfrag,
                                      float* __restrict__ rowmin) {
    __shared__ float2 s_b[BCHUNK * 2];       // 32 KB staged B-fragment chunk

    const int p    = blockIdx.x;
    const int tid  = threadIdx.x;
    const int wave = tid >> 5;
    const int lane = tid & 31;
    const int half = lane >> 4;              // 0: lanes 0-15, 1: lanes 16-31
    const int l16  = lane & 15;
    const int rbase = blockIdx.y * ROWS_PER_BLOCK + wave * 16;

    const float4* ag = Apts + (size_t)p * NPTS;
    const float4* bg = (const float4*)(Bfrag + (size_t)p * NPTS * 2);

    // ---- A fragment (constant for the whole scan) --------------------------
    // ISA 16x4 f32 A layout: lane holds row = l16; lanes 0-15 carry K={0,1},
    // lanes 16-31 carry K={2,3} in VGPR {0,1}.
    float4 av = ag[rbase + l16];
    v2f a;
    a[0] = half ? (-2.f * av.z) : (-2.f * av.x);
    a[1] = half ? 1.f           : (-2.f * av.y);

    // ---- C accumulator = |X_row|^2 per row ---------------------------------
    // C/D layout: VGPR v, lanes 0-15 -> row v, lanes 16-31 -> row v+8.
    v8f cinit;
#pragma unroll
    for (int v = 0; v < 8; ++v)
        cinit[v] = ag[rbase + v + 8 * half].w;

    v8f dmin;
#pragma unroll
    for (int v = 0; v < 8; ++v) dmin[v] = 3.0e38f;

    // ---- scan all B points, 16 columns per WMMA ----------------------------
    for (int cb = 0; cb < NPTS; cb += BCHUNK) {
        // stage chunk (pre-swizzled; copy as float4 for b128 transfers)
        float4* s4 = (float4*)s_b;
        for (int i = tid; i < BCHUNK; i += blockDim.x)
            s4[i] = bg[(cb >> 0) + i];       // 16 B per point, same byte span
        __syncthreads();

        for (int nb = 0; nb < BCHUNK; nb += 16) {
            // Lane's B fragment is directly s_b[nb*2 + lane] (ds_load_b64).
            float2 bf = s_b[nb * 2 + lane];
            v2f b; b[0] = bf.x; b[1] = bf.y;

            v8f d = __builtin_amdgcn_wmma_f32_16x16x4_f32(
                /*neg_a=*/false, a, /*neg_b=*/false, b,
                /*c_mod=*/(short)0, cinit, /*reuse_a=*/false, /*reuse_b=*/false);
#pragma unroll
            for (int v = 0; v < 8; ++v)
                dmin[v] = fminf(dmin[v], d[v]);
        }
        __syncthreads();
    }

    // ---- finish row mins: reduce across the 16 lanes sharing each row ------
#pragma unroll
    for (int v = 0; v < 8; ++v) {
        float r = dmin[v];
        r = fminf(r, __shfl_xor(r, 1));
        r = fminf(r, __shfl_xor(r, 2));
        r = fminf(r, __shfl_xor(r, 4));
        r = fminf(r, __shfl_xor(r, 8));
        if (l16 == 0)
            rowmin[(size_t)p * NPTS + rbase + v + 8 * half] = r;
    }
}

// objective = sum_p w_p * ( mean_m sqrt(rm_cad) + mean_n sqrt(rm_cam) )
__global__ void final_reduce_kernel(const float* __restrict__ w,
                                    const float* __restrict__ rm_cad,
                                    const float* __restrict__ rm_cam,
                                    float* __restrict__ out0) {
    __shared__ float sdata[512];
    const int tid = threadIdx.x;
    float acc = 0.f;
    for (int i = tid; i < PARTS * NPTS; i += blockDim.x) {
        int p = i / NPTS;
        float wp = w[p] * (1.0f / (float)NPTS);
        acc += wp * sqrtf(fmaxf(rm_cad[i], 0.f));
        acc += wp * sqrtf(fmaxf(rm_cam[i], 0.f));
    }
    sdata[tid] = acc;
    __syncthreads();
    for (int s = 256; s > 0; s >>= 1) {
        if (tid < s) sdata[tid] += sdata[tid + s];
        __syncthreads();
    }
    if (tid == 0) out0[0] = sdata[0];
}

extern "C" void kernel_launch(void* const* d_in, const int* in_sizes, int n_in,
                              void* d_out, int out_size, void* d_ws, size_t ws_size,
                              hipStream_t stream) {
    (void)in_sizes; (void)n_in; (void)out_size; (void)ws_size;
    const float* cam  = (const float*)d_in[0];   // (P,N,3)
    const float* cad  = (const float*)d_in[1];   // (P,M,3)
    const float* w    = (const float*)d_in[2];   // (P,)
    const float* quat = (const float*)d_in[3];   // (P,4)
    const float* tra  = (const float*)d_in[4];   // (P,3,1)
    float* out = (float*)d_out;                  // [objective, transforms(P,4,4)]

    // workspace layout
    const size_t PN = (size_t)PARTS * NPTS;
    float4* cadT4    = (float4*)d_ws;            // PN float4
    float4* cam4     = cadT4 + PN;               // PN float4
    float2* bf_cad   = (float2*)(cam4 + PN);     // 2*PN float2 (B-fragment)
    float2* bf_cam   = bf_cad + 2 * PN;          // 2*PN float2
    float*  rm_cad   = (float*)(bf_cam + 2 * PN);
    float*  rm_cam   = rm_cad + PN;

    write_transforms_kernel<<<1, 64, 0, stream>>>(quat, tra, out + 1);
    prep_cad_kernel<<<dim3(PARTS, NPTS / 256), 256, 0, stream>>>(cad, quat, tra, cadT4, bf_cad);
    prep_cam_kernel<<<dim3(PARTS, NPTS / 256), 256, 0, stream>>>(cam, cam4, bf_cam);

    dim3 grid(PARTS, NPTS / ROWS_PER_BLOCK);
    chamfer_rowmin_kernel<<<grid, 512, 0, stream>>>(cadT4, bf_cam, rm_cad); // min over cam per cad
    chamfer_rowmin_kernel<<<grid, 512, 0, stream>>>(cam4, bf_cad, rm_cam); // min over cad per cam

    final_reduce_kernel<<<1, 512, 0, stream>>>(w, rm_cad, rm_cam, out);
}